// TwoHeadsLoss_73667279061691
// MI455X (gfx1250) — compile-verified
//
#include <hip/hip_runtime.h>

typedef __attribute__((ext_vector_type(4))) float v4f;
typedef __attribute__((ext_vector_type(2))) float v2f;
typedef __attribute__((ext_vector_type(8))) float v8f;

#define BLOCKS 2048
#define TPB 256

// ---------------- Stage 1: streaming per-sample loss + block partial sums ----------------
__global__ __launch_bounds__(TPB) void twoheads_loss_partial(
    const float* __restrict__ outputs, const float* __restrict__ targets,
    float* __restrict__ partials, unsigned int n)
{
  const v4f* __restrict__ o4 = reinterpret_cast<const v4f*>(outputs);
  const v4f* __restrict__ t4 = reinterpret_cast<const v4f*>(targets);

  unsigned int g = blockIdx.x * TPB + threadIdx.x;   // chunk id (4 samples/chunk)
  const unsigned int stride = BLOCKS * TPB;

  float acc = 0.0f;
  for (unsigned int c = g; c * 4u < n; c += stride) {
    // Non-temporal: 256MB stream > 192MB L2, don't pollute.
    v4f tt = __builtin_nontemporal_load(&t4[c]);
    v4f oa = __builtin_nontemporal_load(&o4[c * 3u + 0u]);
    v4f ob = __builtin_nontemporal_load(&o4[c * 3u + 1u]);
    v4f oc = __builtin_nontemporal_load(&o4[c * 3u + 2u]);

    // Deinterleave 4 samples of (logit0, logit1, frame)
    float l0[4] = { oa.x, oa.w, ob.z, oc.y };
    float l1[4] = { oa.y, ob.x, ob.w, oc.z };
    float fr[4] = { oa.z, ob.y, oc.x, oc.w };
    float tg[4] = { tt.x, tt.y, tt.z, tt.w };

#pragma unroll
    for (int j = 0; j < 4; ++j) {
      float a = l0[j], b = l1[j], t = tg[j];
      float m   = fmaxf(a, b);
      float lse = m + __logf(__expf(a - m) + __expf(b - m));  // stable 2-way logsumexp
      float ce0 = lse - a;
      float ce1 = lse - b;
      float d   = fr[j] - t;
      float ad  = fabsf(d);
      float sl1 = (ad < 1.0f) ? (0.5f * d * d) : (ad - 0.5f);
      float pos = 0.5f * ce1 + 0.5f * sl1;                    // ALPHA = 0.5
      bool is_neg = (t == -1.0f);
      bool oor    = (t < 0.0f) | (t >= 16.0f);                // MARGIN=0, FRAMES=16
      acc += is_neg ? ce0 : (oor ? 0.0f : pos);
    }
  }

  // wave32 butterfly reduction
#pragma unroll
  for (int off = 16; off >= 1; off >>= 1)
    acc += __shfl_xor(acc, off, 32);

  __shared__ float wsum[TPB / 32];
  const int lane = threadIdx.x & 31;
  const int wid  = threadIdx.x >> 5;
  if (lane == 0) wsum[wid] = acc;
  __syncthreads();
  if (threadIdx.x == 0) {
    float s = 0.0f;
#pragma unroll
    for (int w = 0; w < TPB / 32; ++w) s += wsum[w];
    partials[blockIdx.x] = s;
  }
}

// ---------------- Stage 2: deterministic final reduce; WMMA f32 16x16x4 sums the last 64 ----------------
__global__ __launch_bounds__(256) void twoheads_loss_final(
    const float* __restrict__ partials, float* __restrict__ out, float inv_n)
{
  __shared__ float lds[256];
  __shared__ float red[64];
  __shared__ float rowsum[16];

  const int tid = threadIdx.x;

  // 2048 partials -> 256 per-thread sums (fixed order: deterministic)
  float s = 0.0f;
#pragma unroll
  for (int k = 0; k < BLOCKS / 256; ++k) s += partials[tid + k * 256];
  lds[tid] = s;
  __syncthreads();

  // 256 -> 64, computed uniformly by every thread (no divergence before WMMA)
  const int l64 = tid & 63;
  float v = lds[l64] + lds[l64 + 64] + lds[l64 + 128] + lds[l64 + 192];
  red[l64] = v;          // duplicate writers store identical values: benign
  __syncthreads();

  // 64 -> 16 via V_WMMA_F32_16X16X4_F32 with A == ones.
  // D[m][n] = sum_k B[k][n]; the total over all 64 B slots is invariant to the
  // exact B lane/VGPR mapping, and row 0 of D (VGPR0, lanes 0..15 => N=0..15)
  // holds the 16 column sums. All 8 waves execute this uniformly (EXEC all 1s).
  const int lane = tid & 31;
  v2f bmat; bmat.x = red[lane]; bmat.y = red[lane + 32];
  v2f amat; amat.x = 1.0f;      amat.y = 1.0f;
  v8f cmat = {};
  v8f dmat = __builtin_amdgcn_wmma_f32_16x16x4_f32(
      /*neg_a=*/false, amat, /*neg_b=*/false, bmat,
      /*c_mod=*/(short)0, cmat, /*reuse_a=*/false, /*reuse_b=*/false);

  if (tid < 16) rowsum[tid] = dmat[0];   // lanes 0..15: column sums N=0..15
  __syncthreads();

  if (tid == 0) {
    float tot = 0.0f;
#pragma unroll
    for (int i = 0; i < 16; ++i) tot += rowsum[i];
    out[0] = tot * inv_n;                // mean
  }
}

// ---------------- Launch ----------------
extern "C" void kernel_launch(void* const* d_in, const int* in_sizes, int n_in,
                              void* d_out, int out_size, void* d_ws, size_t ws_size,
                              hipStream_t stream) {
  const float* outputs = (const float*)d_in[0];   // (B, 3) f32
  const float* targets = (const float*)d_in[1];   // (B,)  f32
  unsigned int n = (unsigned int)in_sizes[1];

  float* partials = (float*)d_ws;                 // BLOCKS floats (8 KB)

  twoheads_loss_partial<<<BLOCKS, TPB, 0, stream>>>(outputs, targets, partials, n);
  twoheads_loss_final<<<1, 256, 0, stream>>>(partials, (float*)d_out,
                                             1.0f / (float)n);
}